// Predict_85942295593136
// MI455X (gfx1250) — compile-verified
//
#include <hip/hip_runtime.h>
#include <stdint.h>

typedef unsigned int v4u __attribute__((ext_vector_type(4)));
typedef int          v8i __attribute__((ext_vector_type(8)));
typedef int          v4i __attribute__((ext_vector_type(4)));

#define NT 256
constexpr int   BATCH   = 16;
constexpr int   NANCH   = 25200;      // 1200 + 4800 + 19200
constexpr int   NCLS    = 80;
constexpr int   MAXB    = 150;
constexpr int   CAP     = 4096;       // per-class candidate cap (expected ~912)
constexpr int   FSORT   = 16384;      // finalize sort size (>= 80*150)
constexpr float SCORE_T = 0.9f;
constexpr float IOU_T   = 0.1f;

__constant__ float c_aw[9] = {10.f,16.f,33.f,30.f,62.f,59.f,116.f,156.f,373.f};
__constant__ float c_ah[9] = {13.f,30.f,23.f,61.f,45.f,119.f,90.f,198.f,326.f};

static __device__ __forceinline__ float sigf(float x) { return 1.0f / (1.0f + __expf(-x)); }

// order-preserving float -> u32 (monotone increasing)
static __device__ __forceinline__ uint32_t encf(float f) {
  uint32_t u = __float_as_uint(f);
  return (u & 0x80000000u) ? ~u : (u | 0x80000000u);
}
static __device__ __forceinline__ float decf(uint32_t e) {
  uint32_t u = (e & 0x80000000u) ? (e ^ 0x80000000u) : ~e;
  return __uint_as_float(u);
}

// descending bitonic sort of n (power-of-two) u64 keys in LDS
static __device__ void bitonic_desc(unsigned long long* k, int n, int tid) {
  for (int sz = 2; sz <= n; sz <<= 1) {
    for (int st = sz >> 1; st > 0; st >>= 1) {
      for (int i = tid; i < n; i += NT) {
        int l = i ^ st;
        if (l > i) {
          unsigned long long a = k[i], b = k[l];
          bool doSwap = ((i & sz) == 0) ? (a < b) : (a > b);
          if (doSwap) { k[i] = b; k[l] = a; }
        }
      }
      __syncthreads();
    }
  }
}

// ---------------- Stage A: decode ----------------
__global__ void decode_kernel(const float* __restrict__ p0, const float* __restrict__ p1,
                              const float* __restrict__ p2,
                              float* __restrict__ boxes, float* __restrict__ conf) {
  int t = blockIdx.x * NT + threadIdx.x;
  if (t >= BATCH * NANCH) return;
  int b = t / NANCH, i = t % NANCH;

  const float* fm; int g, loc, ab;
  if (i < 1200)      { fm = p0; g = 20; loc = i;        ab = 6; }
  else if (i < 6000) { fm = p1; g = 40; loc = i - 1200; ab = 3; }
  else               { fm = p2; g = 80; loc = i - 6000; ab = 0; }

  int a = loc % 3, cell = loc / 3;
  int col = cell % g, row = cell / g;
  float ratio = 640.0f / (float)g;

  size_t base = ((size_t)b * (size_t)(g * g * 3) + (size_t)loc) * 85u;
  float t0 = fm[base + 0], t1 = fm[base + 1], t2 = fm[base + 2],
        t3 = fm[base + 3], t4 = fm[base + 4];

  float cx = (sigf(t0) + (float)col) * ratio;
  float cy = (sigf(t1) + (float)row) * ratio;
  float w  = __expf(t2) * c_aw[ab + a];
  float h  = __expf(t3) * c_ah[ab + a];

  float4 bx = make_float4(cx - 0.5f * w, cy - 0.5f * h, cx + 0.5f * w, cy + 0.5f * h);
  ((float4*)boxes)[(size_t)b * NANCH + i] = bx;
  conf[(size_t)b * NANCH + i] = sigf(t4);
}

// ---------------- TDM descriptor helpers ----------------
static __device__ __forceinline__ void set_bits(uint32_t* arr, int lo, int width,
                                                unsigned long long v) {
  for (int i = 0; i < width; ++i) {
    int bp = lo + i;
    uint32_t m = 1u << (bp & 31);
    if ((v >> i) & 1ull) arr[bp >> 5] |= m; else arr[bp >> 5] &= ~m;
  }
}

#if defined(__HIP_DEVICE_COMPILE__) && __has_builtin(__builtin_amdgcn_tensor_load_to_lds)
#define HAVE_TDM 1
// DMA one strided column (element 5+c of each 85-float record) into contiguous LDS.
static __device__ void tdm_gather_col(float* ldsDst, const float* gsrc, int b, int nrows, int c) {
  uint32_t g0[4] = {0, 0, 0, 0};
  uint32_t g1[8] = {0, 0, 0, 0, 0, 0, 0, 0};
  unsigned long long gaddr = (unsigned long long)(uintptr_t)gsrc +
      (((unsigned long long)b * (unsigned long long)nrows * 85ull) + 5ull + (unsigned)c) * 4ull;
  uint32_t laddr = (uint32_t)(uintptr_t)ldsDst;   // low 32 bits of flat = LDS byte offset

  set_bits(g0, 0, 2, 1);                          // count = 1 (valid descriptor)
  set_bits(g0, 32, 32, laddr);                    // lds_addr
  set_bits(g0, 64, 57, gaddr);                    // global_addr (tile start)
  set_bits(g0, 126, 2, 2);                        // type = 2 ("image")

  set_bits(g1, 16, 2, 2);                         // data_size = 4 bytes
  set_bits(g1, 48, 32, 85);                       // tensor_dim0
  set_bits(g1, 80, 32, (unsigned long long)nrows);// tensor_dim1
  set_bits(g1, 112, 16, 1);                       // tile_dim0 = 1 element (4B, ok)
  set_bits(g1, 128, 16, (unsigned long long)nrows);// tile_dim1 = rows
  set_bits(g1, 160, 48, 85);                      // tensor_dim0_stride = 85 elements

  v4u G0 = {g0[0], g0[1], g0[2], g0[3]};
  v8i G1 = {(int)g1[0], (int)g1[1], (int)g1[2], (int)g1[3],
            (int)g1[4], (int)g1[5], (int)g1[6], (int)g1[7]};
  v4i Z4 = {0, 0, 0, 0};                          // groups 2/3 unused (2-D tensor)
  v8i Z8 = {0, 0, 0, 0, 0, 0, 0, 0};              // trailing descriptor group (unused)
  __builtin_amdgcn_tensor_load_to_lds(G0, G1, Z4, Z4, Z8, 0);
}
#else
#define HAVE_TDM 0
#endif

// ---------------- Stage B: per-(image,class) threshold + top-150 + NMS ----------------
__global__ void nms_kernel(const float* __restrict__ p0, const float* __restrict__ p1,
                           const float* __restrict__ p2,
                           const float* __restrict__ confS, const float* __restrict__ boxes,
                           float* __restrict__ topS, int* __restrict__ topI) {
  extern __shared__ float probL[];                // NANCH floats (100800 B, dynamic)
  __shared__ unsigned long long key[CAP];         // 32 KB
  __shared__ float4 cbox[MAXB];
  __shared__ float  carea[MAXB];
  __shared__ uint32_t iouRow[MAXB * 5];
  __shared__ uint32_t keepW[5];
  __shared__ int scnt;

  const int c = blockIdx.x, b = blockIdx.y, tid = threadIdx.x;

  for (int i = tid; i < CAP; i += NT) key[i] = 0ull;
  if (tid == 0) scnt = 0;
  __syncthreads();

#if HAVE_TDM
  if (tid == 0) {   // TDM is wave-level; EXEC ignored -> single issue from wave 0
    tdm_gather_col(probL + 0,    p0, b, 1200,  c);
    tdm_gather_col(probL + 1200, p1, b, 4800,  c);
    tdm_gather_col(probL + 6000, p2, b, 19200, c);
    __builtin_amdgcn_s_wait_tensorcnt(0);
  }
#else
  for (int j = tid; j < NANCH; j += NT) {
    const float* fm; int nf, loc;
    if (j < 1200)      { fm = p0; nf = 1200;  loc = j; }
    else if (j < 6000) { fm = p1; nf = 4800;  loc = j - 1200; }
    else               { fm = p2; nf = 19200; loc = j - 6000; }
    probL[j] = fm[((size_t)b * nf + loc) * 85u + 5u + c];
  }
#endif
  __syncthreads();

  // threshold + compact candidates
  for (int j = tid; j < NANCH; j += NT) {
    float s = confS[(size_t)b * NANCH + j] * sigf(probL[j]);
    if (s > SCORE_T) {
      int p = atomicAdd(&scnt, 1);
      if (p < CAP)
        key[p] = ((unsigned long long)encf(s) << 32) | (uint32_t)(~(uint32_t)j);
    }
  }
  __syncthreads();

  bitonic_desc(key, CAP, tid);                    // desc by score, ties -> lower index

  int K = min(min(scnt, CAP), MAXB);

  if (tid < K) {
    unsigned long long kk = key[tid];
    uint32_t ai = ~(uint32_t)kk;
    float4 bx = ((const float4*)boxes)[(size_t)b * NANCH + ai];
    cbox[tid] = bx;
    carea[tid] = (bx.z - bx.x) * (bx.w - bx.y);
  }
  for (int t2 = tid; t2 < MAXB * 5; t2 += NT) iouRow[t2] = 0;
  __syncthreads();

  // 150x150 IoU > 0.1 bitmask
  for (int t2 = tid; t2 < MAXB * 5; t2 += NT) {
    int i = t2 / 5, w = t2 % 5;
    if (i >= K) continue;
    float4 A = cbox[i]; float aA = carea[i];
    uint32_t bits = 0;
    int j0 = w * 32, j1 = min(j0 + 32, K);
    for (int j = j0; j < j1; ++j) {
      float4 Bx = cbox[j];
      float lx = fmaxf(A.x, Bx.x), ly = fmaxf(A.y, Bx.y);
      float rx = fminf(A.z, Bx.z), ry = fminf(A.w, Bx.w);
      float iw = fmaxf(rx - lx, 0.f), ih = fmaxf(ry - ly, 0.f);
      float inter = iw * ih;
      float iou = inter / (aA + carea[j] - inter + 1e-9f);
      if (iou > IOU_T) bits |= (1u << (j - j0));
    }
    iouRow[i * 5 + w] = bits;
  }
  __syncthreads();

  // greedy NMS over bitmasks (serial, trivial)
  if (tid == 0) {
    uint32_t sup[5] = {0,0,0,0,0}, kp[5] = {0,0,0,0,0};
    for (int i = 0; i < K; ++i) {
      if ((sup[i >> 5] >> (i & 31)) & 1u) continue;
      kp[i >> 5] |= 1u << (i & 31);
      for (int w = 0; w < 5; ++w) {
        uint32_t m = iouRow[i * 5 + w];
        if (w < (i >> 5)) m = 0;
        else if (w == (i >> 5)) m = ((i & 31) == 31) ? 0u : (m & (~0u << ((i & 31) + 1)));
        sup[w] |= m;
      }
    }
    for (int w = 0; w < 5; ++w) keepW[w] = kp[w];
  }
  __syncthreads();

  if (tid < MAXB) {
    bool valid = tid < K;
    float sc = -1.0f; uint32_t ai = 0;
    if (valid) {
      unsigned long long kk = key[tid];
      sc = decf((uint32_t)(kk >> 32));
      ai = ~(uint32_t)kk;
    }
    bool kp = valid && ((keepW[tid >> 5] >> (tid & 31)) & 1u);
    size_t o = ((size_t)b * NCLS + c) * MAXB + tid;
    topS[o] = kp ? sc : -1.0f;                    // fs = where(keep, top_s, -1)
    topI[o] = (int)ai;
  }
}

// ---------------- Stage C: per-image final top-150 ----------------
__global__ void final_kernel(const float* __restrict__ boxes, const float* __restrict__ topS,
                             const int* __restrict__ topI, float* __restrict__ out) {
  extern __shared__ unsigned long long fkey[];    // FSORT keys (128 KB, dynamic)
  const int b = blockIdx.x, tid = threadIdx.x;
  const int NE = NCLS * MAXB;                     // 12000

  for (int e = tid; e < FSORT; e += NT) {
    unsigned long long k = 0ull;
    if (e < NE) {
      float s = topS[(size_t)b * NE + e];
      k = ((unsigned long long)encf(s) << 32) | (uint32_t)(~(uint32_t)e);
    }
    fkey[e] = k;
  }
  __syncthreads();

  bitonic_desc(fkey, FSORT, tid);

  if (tid < MAXB) {
    unsigned long long kk = fkey[tid];
    float s = decf((uint32_t)(kk >> 32));
    uint32_t e = ~(uint32_t)kk;
    bool v = s > 0.0f;                            // kept entries have score > 0.9
    float4 bx = make_float4(-1.f, -1.f, -1.f, -1.f);
    float lab = -1.f, sc = -1.f;
    if (v) {
      int cc = (int)(e / MAXB);
      int anchor = topI[(size_t)b * NE + e];
      bx = ((const float4*)boxes)[(size_t)b * NANCH + anchor];
      lab = (float)cc; sc = s;
    }
    ((float4*)out)[(size_t)b * MAXB + tid] = bx;                       // boxes [B,150,4]
    out[(size_t)BATCH * MAXB * 4 + (size_t)b * MAXB + tid] = sc;       // scores [B,150]
    out[(size_t)BATCH * MAXB * 5 + (size_t)b * MAXB + tid] = lab;      // labels [B,150]
  }
}

extern "C" void kernel_launch(void* const* d_in, const int* in_sizes, int n_in,
                              void* d_out, int out_size, void* d_ws, size_t ws_size,
                              hipStream_t stream) {
  const float* p0 = (const float*)d_in[0];
  const float* p1 = (const float*)d_in[1];
  const float* p2 = (const float*)d_in[2];
  float* out = (float*)d_out;

  float* wsF    = (float*)d_ws;
  float* boxesW = wsF;                                        // B*NANCH*4 floats
  float* confW  = boxesW + (size_t)BATCH * NANCH * 4;         // B*NANCH
  float* topS   = confW  + (size_t)BATCH * NANCH;             // B*NCLS*MAXB
  int*   topI   = (int*)(topS + (size_t)BATCH * NCLS * MAXB); // B*NCLS*MAXB ints

  int tot = BATCH * NANCH;
  decode_kernel<<<(tot + NT - 1) / NT, NT, 0, stream>>>(p0, p1, p2, boxesW, confW);
  nms_kernel<<<dim3(NCLS, BATCH), NT, NANCH * sizeof(float), stream>>>(
      p0, p1, p2, confW, boxesW, topS, topI);
  final_kernel<<<BATCH, NT, FSORT * sizeof(unsigned long long), stream>>>(
      boxesW, topS, topI, out);
}